// Model_36816459661750
// MI455X (gfx1250) — compile-verified
//
#include <hip/hip_runtime.h>

#define H 128

typedef __attribute__((ext_vector_type(2))) float v2f;
typedef __attribute__((ext_vector_type(8))) float v8f;

__device__ __forceinline__ v8f wmma4(v2f a, v2f b, v8f c) {
  // V_WMMA_F32_16X16X4_F32: D = A(16x4) * B(4x16) + C, exact fp32
  return __builtin_amdgcn_wmma_f32_16x16x4_f32(
      /*neg_a=*/false, a, /*neg_b=*/false, b,
      /*c_mod=*/(short)0, c, /*reuse_a=*/false, /*reuse_b=*/false);
}

// ---------------------------------------------------------------------------
// XS[m][n] = sum_k SX[m][k] * W[k][n] + bias[n] + ME[m][n]
// one wave per 16-row slab, 8 col-tiles (full H=128) per wave.
// Guard-free main K loop (rows clamped, stores guarded); tail handles K%4.
// ---------------------------------------------------------------------------
__global__ __launch_bounds__(256) void gemm_xs_kernel(
    const float* __restrict__ SX, const float* __restrict__ W,
    const float* __restrict__ bias, const float* __restrict__ ME,
    float* __restrict__ XS, int M, int K) {
  const int warp = threadIdx.x >> 5;
  const int lane = threadIdx.x & 31;
  const int lo = lane & 15, hi = lane >> 4;
  const int row0 = (blockIdx.x * (blockDim.x >> 5) + warp) * 16;
  const int mc = min(row0 + lo, M - 1);          // clamped row: loads always legal
  const float* __restrict__ arow = SX + (size_t)mc * (size_t)K;

  v8f c[8] = {};
  const int K4 = K & ~3;

#pragma unroll 2
  for (int k = 0; k < K4; k += 4) {
    const int ka = k + 2 * hi;
    const v2f a = *(const v2f*)(arow + ka);                    // b64 load
    const float* __restrict__ w0 = W + (size_t)ka * H + lo;
    const float* __restrict__ w1 = w0 + H;
    if ((k & 63) == 0 && k + 64 < K)
      __builtin_prefetch(arow + ka + 64, 0, 1);                // global_prefetch_b8
#pragma unroll
    for (int t = 0; t < 8; ++t) {
      v2f b;
      b.x = w0[t * 16];
      b.y = w1[t * 16];
      c[t] = wmma4(a, b, c[t]);
    }
  }
  if (K4 < K) {                                   // K%4 tail, guarded once
    const int ka = K4 + 2 * hi;
    v2f a;
    a.x = (ka     < K) ? arow[ka]     : 0.f;
    a.y = (ka + 1 < K) ? arow[ka + 1] : 0.f;
#pragma unroll
    for (int t = 0; t < 8; ++t) {
      const int col = t * 16 + lo;
      v2f b;
      b.x = (ka     < K) ? W[(size_t)ka * H + col]       : 0.f;
      b.y = (ka + 1 < K) ? W[(size_t)(ka + 1) * H + col] : 0.f;
      c[t] = wmma4(a, b, c[t]);
    }
  }

#pragma unroll
  for (int t = 0; t < 8; ++t)
#pragma unroll
    for (int r = 0; r < 8; ++r) {
      const int row = row0 + r + 8 * hi;
      const int col = t * 16 + lo;
      if (row < M)
        XS[(size_t)row * H + col] = c[t][r] + bias[col] + ME[(size_t)row * H + col];
    }
}

// ---------------------------------------------------------------------------
// OUT[m] = (AGG[m]/max(cnt[m],1)) @ WL + X[m] @ WR + BL   (optional relu)
// ---------------------------------------------------------------------------
__global__ __launch_bounds__(256) void gemm_sage_kernel(
    const float* __restrict__ AGG, const float* __restrict__ CNT,
    const float* __restrict__ X,
    const float* __restrict__ WL, const float* __restrict__ WR,
    const float* __restrict__ BL, float* __restrict__ OUT, int M, int relu) {
  const int warp = threadIdx.x >> 5;
  const int lane = threadIdx.x & 31;
  const int lo = lane & 15, hi = lane >> 4;
  const int row0 = (blockIdx.x * (blockDim.x >> 5) + warp) * 16;
  const int mc = min(row0 + lo, M - 1);
  const float inv = 1.f / fmaxf(CNT[mc], 1.f);     // mean fused into A frag
  const float* __restrict__ ag = AGG + (size_t)mc * H;
  const float* __restrict__ xr = X   + (size_t)mc * H;

  v8f c[8] = {};
#pragma unroll 2
  for (int k = 0; k < H; k += 4) {
    const int ka = k + 2 * hi;
    v2f a = *(const v2f*)(ag + ka);
    a.x *= inv;
    a.y *= inv;
    const float* __restrict__ w0 = WL + (size_t)ka * H + lo;
    const float* __restrict__ w1 = w0 + H;
#pragma unroll
    for (int t = 0; t < 8; ++t) {
      v2f b;
      b.x = w0[t * 16];
      b.y = w1[t * 16];
      c[t] = wmma4(a, b, c[t]);
    }
  }
#pragma unroll 2
  for (int k = 0; k < H; k += 4) {
    const int ka = k + 2 * hi;
    const v2f a = *(const v2f*)(xr + ka);
    const float* __restrict__ w0 = WR + (size_t)ka * H + lo;
    const float* __restrict__ w1 = w0 + H;
#pragma unroll
    for (int t = 0; t < 8; ++t) {
      v2f b;
      b.x = w0[t * 16];
      b.y = w1[t * 16];
      c[t] = wmma4(a, b, c[t]);
    }
  }

#pragma unroll
  for (int t = 0; t < 8; ++t)
#pragma unroll
    for (int r = 0; r < 8; ++r) {
      const int row = row0 + r + 8 * hi;
      const int col = t * 16 + lo;
      if (row < M) {
        float v = c[t][r] + BL[col];
        OUT[(size_t)row * H + col] = relu ? fmaxf(v, 0.f) : v;
      }
    }
}

// ---------------------------------------------------------------------------
// P[m][j] = sum_n A[m][n] * W[j][n]     (B transposed; A is MxH, W is HxH)
// ---------------------------------------------------------------------------
__global__ __launch_bounds__(256) void gemm_bt_kernel(
    const float* __restrict__ A, const float* __restrict__ W,
    float* __restrict__ P, int M) {
  const int warp = threadIdx.x >> 5;
  const int lane = threadIdx.x & 31;
  const int lo = lane & 15, hi = lane >> 4;
  const int row0 = (blockIdx.x * (blockDim.x >> 5) + warp) * 16;
  const int mc = min(row0 + lo, M - 1);
  const float* __restrict__ ar = A + (size_t)mc * H;

  v8f c[8] = {};
#pragma unroll 2
  for (int k = 0; k < H; k += 4) {
    const int ka = k + 2 * hi;
    const v2f a = *(const v2f*)(ar + ka);
#pragma unroll
    for (int t = 0; t < 8; ++t) {
      const int col = t * 16 + lo;
      const v2f b = *(const v2f*)(W + (size_t)col * H + ka);   // B(r,j)=W[j][r], b64 load
      c[t] = wmma4(a, b, c[t]);
    }
  }
#pragma unroll
  for (int t = 0; t < 8; ++t)
#pragma unroll
    for (int r = 0; r < 8; ++r) {
      const int row = row0 + r + 8 * hi;
      if (row < M) P[(size_t)row * H + t * 16 + lo] = c[t][r];
    }
}

// ---------------------------------------------------------------------------
// per-edge degree counts (both directions in one pass)
// ---------------------------------------------------------------------------
__global__ void count_kernel(const int* __restrict__ src_u,
                             const int* __restrict__ dst_s,
                             float* __restrict__ cnt_u,
                             float* __restrict__ cnt_s, int E) {
  int e = blockIdx.x * blockDim.x + threadIdx.x;
  if (e < E) {
    atomicAdd(&cnt_s[dst_s[e]], 1.f);
    atomicAdd(&cnt_u[src_u[e]], 1.f);
  }
}

// ---------------------------------------------------------------------------
// AGG[didx[e]] += FEAT[sidx[e]]  — one wave per edge, 4 ch per lane
// ---------------------------------------------------------------------------
__global__ void scatter_kernel(const float* __restrict__ feat,
                               const int* __restrict__ sidx,
                               const int* __restrict__ didx,
                               float* __restrict__ agg, int E) {
  int e = blockIdx.x * (blockDim.x >> 5) + (threadIdx.x >> 5);
  int lane = threadIdx.x & 31;
  if (e < E) {
    int s = sidx[e], d = didx[e];
    float4 v = ((const float4*)(feat + (size_t)s * H))[lane];
    float* dp = agg + (size_t)d * H + lane * 4;
    atomicAdd(dp + 0, v.x);
    atomicAdd(dp + 1, v.y);
    atomicAdd(dp + 2, v.z);
    atomicAdd(dp + 3, v.w);
  }
}

// ---------------------------------------------------------------------------
// out[l] = mean2_u[i]·P1[j] + h_u[i]·P2[j] + bl2·o_s[j]
// (algebraic factorization of o_u[i]·o_s[j]; one wave per label edge)
// ---------------------------------------------------------------------------
__global__ void classifier_kernel(const int* __restrict__ eli,
                                  const int* __restrict__ elj,
                                  const float* __restrict__ AGGU,
                                  const float* __restrict__ CNTU,
                                  const float* __restrict__ HU,
                                  const float* __restrict__ P1,
                                  const float* __restrict__ P2,
                                  const float* __restrict__ OS,
                                  const float* __restrict__ BL2,
                                  float* __restrict__ out, int L) {
  int l = blockIdx.x * (blockDim.x >> 5) + (threadIdx.x >> 5);
  int lane = threadIdx.x & 31;
  if (l >= L) return;  // wave-uniform
  int i = eli[l], j = elj[l];
  float inv = 1.f / fmaxf(CNTU[i], 1.f);
  float4 mu = ((const float4*)(AGGU + (size_t)i * H))[lane];
  float4 p1 = ((const float4*)(P1 + (size_t)j * H))[lane];
  float4 hu = ((const float4*)(HU + (size_t)i * H))[lane];
  float4 p2 = ((const float4*)(P2 + (size_t)j * H))[lane];
  float4 os = ((const float4*)(OS + (size_t)j * H))[lane];
  float4 bl = ((const float4*)BL2)[lane];
  float acc = inv * (mu.x * p1.x + mu.y * p1.y + mu.z * p1.z + mu.w * p1.w) +
              (hu.x * p2.x + hu.y * p2.y + hu.z * p2.z + hu.w * p2.w) +
              (bl.x * os.x + bl.y * os.y + bl.z * os.z + bl.w * os.w);
#pragma unroll
  for (int off = 16; off > 0; off >>= 1) acc += __shfl_xor(acc, off, 32);
  if (lane == 0) out[l] = acc;
}

// ---------------------------------------------------------------------------
extern "C" void kernel_launch(void* const* d_in, const int* in_sizes, int n_in,
                              void* d_out, int out_size, void* d_ws, size_t ws_size,
                              hipStream_t stream) {
  const int NUSER = in_sizes[0];
  const int NSUB = in_sizes[1];
  const float* SX = (const float*)d_in[2];
  const int F = in_sizes[2] / NSUB;
  const int* EI = (const int*)d_in[3];
  const int E = in_sizes[3] / 2;
  const int* src_u = EI;
  const int* dst_s = EI + E;
  const int* ELI = (const int*)d_in[4];
  const int L = in_sizes[4] / 2;
  const float* user_emb  = (const float*)d_in[5];
  const float* movie_emb = (const float*)d_in[6];
  const float* mw = (const float*)d_in[7];
  const float* mb = (const float*)d_in[8];
  const float* w1us_l = (const float*)d_in[9];
  const float* b1us   = (const float*)d_in[10];
  const float* w1us_r = (const float*)d_in[11];
  const float* w1su_l = (const float*)d_in[12];
  const float* b1su   = (const float*)d_in[13];
  const float* w1su_r = (const float*)d_in[14];
  const float* w2us_l = (const float*)d_in[15];
  const float* b2us   = (const float*)d_in[16];
  const float* w2us_r = (const float*)d_in[17];
  const float* w2su_l = (const float*)d_in[18];
  const float* b2su   = (const float*)d_in[19];
  const float* w2su_r = (const float*)d_in[20];

  // --- workspace carve (256B aligned) ---
  char* p = (char*)d_ws;
  auto carve = [&](size_t bytes) -> float* {
    float* r = (float*)p;
    p += (bytes + 255) & ~(size_t)255;
    return r;
  };
  const size_t szS = (size_t)NSUB * H * sizeof(float);
  const size_t szU = (size_t)NUSER * H * sizeof(float);
  float* XS   = carve(szS);              // x_s, later reused as o_s
  float* HS   = carve(szS);
  float* AGGS = carve(szS);
  float* P1   = carve(szS);
  float* P2   = carve(szS);
  float* CNTS = carve((size_t)NSUB * sizeof(float));
  float* HU   = carve(szU);
  float* AGGU = carve(szU);
  float* CNTU = carve((size_t)NUSER * sizeof(float));
  (void)ws_size; (void)n_in;

  hipMemsetAsync(AGGS, 0, szS, stream);
  hipMemsetAsync(AGGU, 0, szU, stream);
  hipMemsetAsync(CNTS, 0, (size_t)NSUB * sizeof(float), stream);
  hipMemsetAsync(CNTU, 0, (size_t)NUSER * sizeof(float), stream);

  const int WPB = 8;                       // waves per block (256 threads)
  dim3 blk(256);
  int gS = (NSUB + 16 * WPB - 1) / (16 * WPB);
  int gU = (NUSER + 16 * WPB - 1) / (16 * WPB);
  int gE = (E + WPB - 1) / WPB;
  int gL = (L + WPB - 1) / WPB;

  // x_s = subreddit_x @ movie_lin_w + b + movie_emb
  gemm_xs_kernel<<<gS, blk, 0, stream>>>(SX, mw, mb, movie_emb, XS, NSUB, F);
  // degrees (shared by both layers)
  count_kernel<<<(E + 255) / 256, 256, 0, stream>>>(src_u, dst_s, CNTU, CNTS, E);
  // layer 1 u->s
  scatter_kernel<<<gE, blk, 0, stream>>>(user_emb, src_u, dst_s, AGGS, E);
  gemm_sage_kernel<<<gS, blk, 0, stream>>>(AGGS, CNTS, XS, w1us_l, w1us_r, b1us, HS, NSUB, 1);
  // layer 1 s->u
  scatter_kernel<<<gE, blk, 0, stream>>>(XS, dst_s, src_u, AGGU, E);
  gemm_sage_kernel<<<gU, blk, 0, stream>>>(AGGU, CNTU, user_emb, w1su_l, w1su_r, b1su, HU, NUSER, 1);
  // layer 2 u->s  (o_s written into XS, which is now free)
  hipMemsetAsync(AGGS, 0, szS, stream);
  scatter_kernel<<<gE, blk, 0, stream>>>(HU, src_u, dst_s, AGGS, E);
  gemm_sage_kernel<<<gS, blk, 0, stream>>>(AGGS, CNTS, HS, w2us_l, w2us_r, b2us, XS, NSUB, 0);
  // layer 2 s->u aggregation only (o_u never materialized)
  hipMemsetAsync(AGGU, 0, szU, stream);
  scatter_kernel<<<gE, blk, 0, stream>>>(HS, dst_s, src_u, AGGU, E);
  // P1 = o_s @ WL2^T, P2 = o_s @ WR2^T
  gemm_bt_kernel<<<gS, blk, 0, stream>>>(XS, w2su_l, P1, NSUB);
  gemm_bt_kernel<<<gS, blk, 0, stream>>>(XS, w2su_r, P2, NSUB);
  // classifier
  classifier_kernel<<<gL, blk, 0, stream>>>(ELI, ELI + L, AGGU, CNTU, HU, P1, P2,
                                            XS, b2su, (float*)d_out, L);
}